// GR_39359080301143
// MI455X (gfx1250) — compile-verified
//
#include <hip/hip_runtime.h>
#include <math.h>

// ---------------------------------------------------------------------------
// MI455X / gfx1250 implementation using V_WMMA_F32_16X16X4_F32 (full fp32
// matrix path; wave32). All GEMMs tiled 16x16 per wave; operand layouts per
// CDNA5 ISA 7.12.2:
//   A 16x4 f32: lane L -> row m = L%16, K pair at kb = (L/16)*2
//   B 4x16 f32: lane L -> col n = L%16, K pair at kb = (L/16)*2
//   C/D 16x16 : lane L -> col n = L%16, VGPR r -> row r + 8*(L/16)
// ---------------------------------------------------------------------------

typedef float v2f __attribute__((ext_vector_type(2)));
typedef float v8f __attribute__((ext_vector_type(8)));

constexpr int B  = 8;
constexpr int C  = 256;
constexpr int H  = 48;
constexpr int W  = 48;
constexpr int HW = H * W;      // 2304
constexpr int M  = C / 2;      // 128
constexpr int NT = HW / 16;    // 144 tiles along HW

__device__ __forceinline__ float sigmoidf_(float x) {
    return 1.0f / (1.0f + __expf(-x));
}

__device__ __forceinline__ v8f wmma4(v2f a, v2f b, v8f c) {
    // D = A(16x4) * B(4x16) + C, fp32 in/out
    return __builtin_amdgcn_wmma_f32_16x16x4_f32(
        /*neg_a=*/false, a, /*neg_b=*/false, b,
        /*c_mod=*/(short)0, c, /*reuse_a=*/false, /*reuse_b=*/false);
}

// ---------------------------------------------------------------------------
// K1: per-(b,c) mean and max over HW
// ---------------------------------------------------------------------------
__global__ void avgmax_kernel(const float* __restrict__ ftr,
                              float* __restrict__ avg, float* __restrict__ mx) {
    int bc = blockIdx.x;                       // [0, B*C)
    const float* p = ftr + (size_t)bc * HW;
    float s = 0.f, m = -3.402823466e38f;
    for (int i = threadIdx.x; i < HW; i += blockDim.x) {
        float v = p[i];
        s += v;
        m = fmaxf(m, v);
    }
    __shared__ float ss[256], sm[256];
    ss[threadIdx.x] = s; sm[threadIdx.x] = m;
    __syncthreads();
    for (int off = 128; off > 0; off >>= 1) {
        if ((int)threadIdx.x < off) {
            ss[threadIdx.x] += ss[threadIdx.x + off];
            sm[threadIdx.x] = fmaxf(sm[threadIdx.x], sm[threadIdx.x + off]);
        }
        __syncthreads();
    }
    if (threadIdx.x == 0) {
        avg[bc] = ss[0] * (1.0f / (float)HW);
        mx[bc]  = sm[0];
    }
}

// ---------------------------------------------------------------------------
// K2: cw[b,m] = sigmoid(relu(avg·Wa[m]) + relu(mx·Wm[m]))   (tiny)
// ---------------------------------------------------------------------------
__global__ void cw_kernel(const float* __restrict__ avg, const float* __restrict__ mx,
                          const float* __restrict__ wa, const float* __restrict__ wm,
                          float* __restrict__ cw) {
    int t = blockIdx.x * blockDim.x + threadIdx.x;
    if (t >= B * M) return;
    int b = t / M, m = t % M;
    const float* av = avg + b * C;
    const float* mv = mx + b * C;
    const float* ra = wa + m * C;
    const float* rm = wm + m * C;
    float s1 = 0.f, s2 = 0.f;
    for (int c = 0; c < C; ++c) { s1 += av[c] * ra[c]; s2 += mv[c] * rm[c]; }
    cw[t] = sigmoidf_(fmaxf(s1, 0.f) + fmaxf(s2, 0.f));
}

// ---------------------------------------------------------------------------
// K3: k[b,m,n] = relu(W_k @ x + bias);  kq = k * cw  (both stored [B,M,HW])
// GEMM: rows m (A = conv_k_w [M,C]), cols n (B = x [C,HW] row-major, coalesced)
// ---------------------------------------------------------------------------
__global__ void k_kernel(const float* __restrict__ ftr, const float* __restrict__ wk,
                         const float* __restrict__ bk, const float* __restrict__ cw,
                         float* __restrict__ kbuf, float* __restrict__ kqbuf) {
    const int wave = threadIdx.x >> 5, lane = threadIdx.x & 31;
    const int col = lane & 15, half = lane >> 4;
    int bt = blockIdx.x;                       // B * 8 * (NT/4)
    const int ntg = bt % (NT / 4); bt /= (NT / 4);
    const int mt = bt % 8;
    const int b  = bt / 8;
    const int m0 = mt * 16, n0 = (ntg * 4 + wave) * 16;

    const float* x = ftr + (size_t)b * C * HW;
    const int arow = m0 + col;
    v8f acc = {};
    for (int k0 = 0; k0 < C; k0 += 4) {
        const int kk = k0 + 2 * half;
        v2f a;  a.x = wk[arow * C + kk];             a.y = wk[arow * C + kk + 1];
        v2f bf; bf.x = x[(size_t)kk * HW + n0 + col]; bf.y = x[(size_t)(kk + 1) * HW + n0 + col];
        acc = wmma4(a, bf, acc);
    }
    const size_t base = (size_t)b * M * HW;
#pragma unroll
    for (int r = 0; r < 8; ++r) {
        const int m = m0 + r + 8 * half;
        const float v = fmaxf(acc[r] + bk[m], 0.f);
        kbuf [base + (size_t)m * HW + n0 + col] = v;
        kqbuf[base + (size_t)m * HW + n0 + col] = v * cw[b * M + m];
    }
}

// ---------------------------------------------------------------------------
// K4 (dominant): d[b,n] = rsqrt( sum_p sigmoid( sum_m kq[m,n]*k[m,p] ) )
// S never materialized. Per block: one 16-row strip of S; 4 waves split the
// 144 p-tiles. A-frags (kq^T strip, K=128) cached in 64 VGPRs across p-loop.
// ---------------------------------------------------------------------------
__global__ void d_kernel(const float* __restrict__ kbuf, const float* __restrict__ kqbuf,
                         float* __restrict__ dbuf) {
    const int wave = threadIdx.x >> 5, lane = threadIdx.x & 31;
    const int col = lane & 15, half = lane >> 4;
    const int nt = blockIdx.x % NT, b = blockIdx.x / NT;
    const int n0 = nt * 16;
    const float* kq = kqbuf + (size_t)b * M * HW;
    const float* kp = kbuf + (size_t)b * M * HW;

    __shared__ float rowsum[16];
    if (threadIdx.x < 16) rowsum[threadIdx.x] = 0.f;
    __syncthreads();

    // Preload A fragments: A[n][K=m] = kq[m*HW + n]  (K-major -> coalesced)
    v2f av[32];
#pragma unroll
    for (int s = 0; s < 32; ++s) {
        const int kk = 4 * s + 2 * half;
        av[s].x = kq[(size_t)kk * HW + n0 + col];
        av[s].y = kq[(size_t)(kk + 1) * HW + n0 + col];
    }

    float rs[8] = {0.f, 0.f, 0.f, 0.f, 0.f, 0.f, 0.f, 0.f};
    for (int pt = wave; pt < NT; pt += 4) {
        const int p0 = pt * 16;
        v8f acc = {};
#pragma unroll
        for (int s = 0; s < 32; ++s) {
            const int kk = 4 * s + 2 * half;
            v2f bf;
            bf.x = kp[(size_t)kk * HW + p0 + col];
            bf.y = kp[(size_t)(kk + 1) * HW + p0 + col];
            acc = wmma4(av[s], bf, acc);
        }
#pragma unroll
        for (int r = 0; r < 8; ++r) rs[r] += sigmoidf_(acc[r]);
    }
    // Reduce over 16 columns (lanes within each half) per row.
#pragma unroll
    for (int r = 0; r < 8; ++r)
        for (int off = 1; off < 16; off <<= 1)
            rs[r] += __shfl_xor(rs[r], off, 32);
    if (col == 0) {
#pragma unroll
        for (int r = 0; r < 8; ++r) atomicAdd(&rowsum[r + 8 * half], rs[r]);
    }
    __syncthreads();
    if (threadIdx.x < 16)
        dbuf[b * HW + n0 + threadIdx.x] = __frsqrt_rn(rowsum[threadIdx.x]);
}

// ---------------------------------------------------------------------------
// K5: mid[b,m,c] = cw[m] * sum_n (d[n]*k[m,n]) * ftr[c,n]   ("NT" GEMM, K=HW)
// ---------------------------------------------------------------------------
__global__ void mid_kernel(const float* __restrict__ ftr, const float* __restrict__ kbuf,
                           const float* __restrict__ dbuf, const float* __restrict__ cw,
                           float* __restrict__ mid) {
    const int wave = threadIdx.x >> 5, lane = threadIdx.x & 31;
    const int col = lane & 15, half = lane >> 4;
    int bt = blockIdx.x;                        // B * 8 * 4
    const int ctg = bt % 4; bt /= 4;
    const int mt = bt % 8;
    const int b  = bt / 8;
    const int m0 = mt * 16, c0 = (ctg * 4 + wave) * 16;

    const float* kp = kbuf + (size_t)b * M * HW;
    const float* x  = ftr + (size_t)b * C * HW;
    const float* d  = dbuf + b * HW;
    const int arow = m0 + col;   // A row m
    const int bcol = c0 + col;   // B col c
    v8f acc = {};
    for (int k0 = 0; k0 < HW; k0 += 4) {
        const int kk = k0 + 2 * half;
        v2f a;  a.x = kp[(size_t)arow * HW + kk] * d[kk];
                a.y = kp[(size_t)arow * HW + kk + 1] * d[kk + 1];
        v2f bf; bf.x = x[(size_t)bcol * HW + kk];
                bf.y = x[(size_t)bcol * HW + kk + 1];
        acc = wmma4(a, bf, acc);
    }
#pragma unroll
    for (int r = 0; r < 8; ++r) {
        const int m = m0 + r + 8 * half;
        mid[((size_t)b * M + m) * C + c0 + col] = acc[r] * cw[b * M + m];
    }
}

// ---------------------------------------------------------------------------
// K6: LX^T[b,c,n] = ftr[c,n] - d[n] * sum_m mid[m,c]*k[m,n]
// Stored transposed [B,C,HW] so K6 and K7 both stream fully coalesced.
// ---------------------------------------------------------------------------
__global__ void lx_kernel(const float* __restrict__ ftr, const float* __restrict__ kbuf,
                          const float* __restrict__ mid, const float* __restrict__ dbuf,
                          float* __restrict__ lxt) {
    const int wave = threadIdx.x >> 5, lane = threadIdx.x & 31;
    const int col = lane & 15, half = lane >> 4;
    int bt = blockIdx.x;                        // B * 16 * (NT/4)
    const int ntg = bt % (NT / 4); bt /= (NT / 4);
    const int ct = bt % 16;
    const int b  = bt / 16;
    const int c0 = ct * 16, n0 = (ntg * 4 + wave) * 16;

    const float* kp = kbuf + (size_t)b * M * HW;
    const float* mp = mid + (size_t)b * M * C;
    v8f acc = {};
    for (int k0 = 0; k0 < M; k0 += 4) {
        const int kk = k0 + 2 * half;
        v2f a;  a.x = mp[kk * C + c0 + col];          a.y = mp[(kk + 1) * C + c0 + col];
        v2f bf; bf.x = kp[(size_t)kk * HW + n0 + col]; bf.y = kp[(size_t)(kk + 1) * HW + n0 + col];
        acc = wmma4(a, bf, acc);
    }
    const float dn = dbuf[b * HW + n0 + col];
    const float* x = ftr + (size_t)b * C * HW;
#pragma unroll
    for (int r = 0; r < 8; ++r) {
        const int c = c0 + r + 8 * half;
        lxt[((size_t)b * C + c) * HW + n0 + col] =
            x[(size_t)c * HW + n0 + col] - dn * acc[r];
    }
}

// ---------------------------------------------------------------------------
// K7: out[b,c,n] = ftr[c,n] + sum_c' gcn_w[c',c] * LX^T[c',n]
// (Y transposed back to [B,C,H,W] for free.)
// ---------------------------------------------------------------------------
__global__ void y_kernel(const float* __restrict__ ftr, const float* __restrict__ lxt,
                         const float* __restrict__ gcn, float* __restrict__ out) {
    const int wave = threadIdx.x >> 5, lane = threadIdx.x & 31;
    const int col = lane & 15, half = lane >> 4;
    int bt = blockIdx.x;                        // B * 16 * (NT/4)
    const int ntg = bt % (NT / 4); bt /= (NT / 4);
    const int ct = bt % 16;
    const int b  = bt / 16;
    const int c0 = ct * 16, n0 = (ntg * 4 + wave) * 16;

    const float* lp = lxt + (size_t)b * C * HW;
    v8f acc = {};
    for (int k0 = 0; k0 < C; k0 += 4) {
        const int kk = k0 + 2 * half;
        v2f a;  a.x = gcn[kk * C + c0 + col];          a.y = gcn[(kk + 1) * C + c0 + col];
        v2f bf; bf.x = lp[(size_t)kk * HW + n0 + col];  bf.y = lp[(size_t)(kk + 1) * HW + n0 + col];
        acc = wmma4(a, bf, acc);
    }
    const float* x = ftr + (size_t)b * C * HW;
#pragma unroll
    for (int r = 0; r < 8; ++r) {
        const int c = c0 + r + 8 * half;
        out[((size_t)b * C + c) * HW + n0 + col] =
            x[(size_t)c * HW + n0 + col] + acc[r];
    }
}

// ---------------------------------------------------------------------------
extern "C" void kernel_launch(void* const* d_in, const int* in_sizes, int n_in,
                              void* d_out, int out_size, void* d_ws, size_t ws_size,
                              hipStream_t stream) {
    (void)in_sizes; (void)n_in; (void)out_size; (void)ws_size;
    const float* ftr      = (const float*)d_in[0];
    const float* conv_k_w = (const float*)d_in[1];
    const float* conv_k_b = (const float*)d_in[2];
    const float* avg_fc_w = (const float*)d_in[3];
    const float* max_fc_w = (const float*)d_in[4];
    const float* gcn_w    = (const float*)d_in[5];
    float* out = (float*)d_out;

    float* ws    = (float*)d_ws;
    float* avg   = ws;                                  // B*C
    float* mx    = avg + B * C;                         // B*C
    float* cw    = mx + B * C;                          // B*M
    float* dbuf  = cw + B * M;                          // B*HW
    float* kbuf  = dbuf + B * HW;                       // B*M*HW
    float* kqbuf = kbuf + (size_t)B * M * HW;           // B*M*HW
    float* mid   = kqbuf + (size_t)B * M * HW;          // B*M*C
    float* lxt   = mid + (size_t)B * M * C;             // B*C*HW

    hipLaunchKernelGGL(avgmax_kernel, dim3(B * C), dim3(256), 0, stream, ftr, avg, mx);
    hipLaunchKernelGGL(cw_kernel, dim3((B * M + 255) / 256), dim3(256), 0, stream,
                       avg, mx, avg_fc_w, max_fc_w, cw);
    hipLaunchKernelGGL(k_kernel, dim3(B * 8 * (NT / 4)), dim3(128), 0, stream,
                       ftr, conv_k_w, conv_k_b, cw, kbuf, kqbuf);
    hipLaunchKernelGGL(d_kernel, dim3(B * NT), dim3(128), 0, stream, kbuf, kqbuf, dbuf);
    hipLaunchKernelGGL(mid_kernel, dim3(B * 8 * 4), dim3(128), 0, stream,
                       ftr, kbuf, dbuf, cw, mid);
    hipLaunchKernelGGL(lx_kernel, dim3(B * 16 * (NT / 4)), dim3(128), 0, stream,
                       ftr, kbuf, mid, dbuf, lxt);
    hipLaunchKernelGGL(y_kernel, dim3(B * 16 * (NT / 4)), dim3(128), 0, stream,
                       ftr, lxt, gcn_w, out);
}